// AssistantGenerator_69870527971906
// MI455X (gfx1250) — compile-verified
//
#include <hip/hip_runtime.h>
#include <hip/hip_bf16.h>
#include <cfloat>

// Problem constants (match the reference)
#define BB 16
#define LL 128
#define RR 200
#define RP 208           // R padded to a multiple of 16 (13 N-tiles)
#define HH 768
#define VV 30522

typedef __attribute__((ext_vector_type(2))) float v2f;
typedef __attribute__((ext_vector_type(8))) float v8f;

// ---------------------------------------------------------------------------
// Kernel 0: zero-fill the 250 MB output with wide stores (bandwidth floor).
// ---------------------------------------------------------------------------
__global__ void zero_out_kernel(float4* __restrict__ out, long n4) {
    long i = (long)blockIdx.x * blockDim.x + threadIdx.x;
    long stride = (long)gridDim.x * blockDim.x;
    float4 z = make_float4(0.f, 0.f, 0.f, 0.f);
    for (; i < n4; i += stride) out[i] = z;
}

// ---------------------------------------------------------------------------
// Kernel 1: Y[padRows, H] = X[validRows, H] @ W[H, H] + bias, f32 WMMA.
// One wave (32 threads) per 16x16 output tile; K-loop of 4 per WMMA.
// Rows beyond validPerBatch (k-matrix padding) read row 0 and are later
// masked out in softmax, so their contents are irrelevant.
// ---------------------------------------------------------------------------
__global__ __launch_bounds__(32)
void gemm_proj_kernel(const float* __restrict__ X, const float* __restrict__ W,
                      const float* __restrict__ bias, float* __restrict__ Y,
                      int padPerBatch, int validPerBatch) {
    const int tilesN = HH / 16;                // 48
    int tileM = blockIdx.x / tilesN;
    int tileN = blockIdx.x % tilesN;
    int lane  = threadIdx.x;                   // wave32
    int half  = lane >> 4;                     // K-group select
    int m15   = lane & 15;
    int col   = tileN * 16 + m15;

    // Map padded output row -> source row (clamp padding rows to batch row 0)
    int padRow = tileM * 16 + m15;
    int b = padRow / padPerBatch;
    int r = padRow % padPerBatch;
    int srcRow = b * validPerBatch + ((r < validPerBatch) ? r : 0);

    const float* xrow = X + (size_t)srcRow * HH;
    v8f c = {};
    for (int k0 = 0; k0 < HH; k0 += 4) {
        int kk = k0 + half * 2;
        v2f a = *(const v2f*)(xrow + kk);           // A: contiguous K
        v2f bm;
        bm.x = W[(size_t)kk * HH + col];            // B: W[k][col], stride H
        bm.y = W[(size_t)(kk + 1) * HH + col];
        c = __builtin_amdgcn_wmma_f32_16x16x4_f32(false, a, false, bm,
                                                  (short)0, c, false, false);
    }
    float bv = bias[col];
#pragma unroll
    for (int i = 0; i < 8; ++i) {
        int row = tileM * 16 + i + half * 8;        // C/D layout
        Y[(size_t)row * HH + col] = c[i] + bv;
    }
}

// ---------------------------------------------------------------------------
// Kernel 2: fused scores (WMMA q.k^T) -> masked softmax -> vocab scatter.
// One block = one (batch, 16-row L-tile). 8 waves cover the 13 R-tiles;
// scores live in LDS, only scatter writes touch HBM.
// ---------------------------------------------------------------------------
__global__ __launch_bounds__(256)
void attn_scatter_kernel(const float* __restrict__ qws,
                         const float* __restrict__ kws,
                         const int* __restrict__ ids,
                         const unsigned char* __restrict__ mask,
                         float* __restrict__ out) {
    __shared__ float sc[16][RP];      // 13.3 KB score tile
    __shared__ float red[16][16];
    __shared__ float rowmax[16];
    __shared__ float rowsum[16];

    int b  = blockIdx.x / (LL / 16);
    int l0 = (blockIdx.x % (LL / 16)) * 16;

    int tid  = threadIdx.x;
    int wave = tid >> 5;              // uniform within a wave
    int lane = tid & 31;
    int half = lane >> 4;
    int m15  = lane & 15;
    const float scale = 0.03608439182435161f;   // 1/sqrt(768)

    // ---- Phase 1: scores[16 x 208] via WMMA (wave-uniform tile loop) ----
    for (int t = wave; t < RP / 16; t += 8) {
        int n = t * 16 + m15;
        const float* qrow = qws + (size_t)(b * LL + l0 + m15) * HH;
        const float* krow = kws + (size_t)(b * RP + n) * HH;
        v8f c = {};
        for (int k0 = 0; k0 < HH; k0 += 4) {
            int kk = k0 + half * 2;
            v2f a  = *(const v2f*)(qrow + kk);   // A[M=m15][K]
            v2f bm = *(const v2f*)(krow + kk);   // B[K][N=n] = k[n][K]
            c = __builtin_amdgcn_wmma_f32_16x16x4_f32(false, a, false, bm,
                                                      (short)0, c, false, false);
        }
#pragma unroll
        for (int i = 0; i < 8; ++i)
            sc[i + half * 8][n] = c[i] * scale;
    }
    __syncthreads();

    // ---- Phase 2: masked softmax over R (16 threads per row) ----
    int g = tid >> 4;                 // row within L-tile
    int j = tid & 15;

    float pmax = -FLT_MAX;
    for (int t = 0; t < RP / 16; ++t) {
        int r = j + t * 16;
        if (r < RR) {
            float v = mask[b * RR + r] ? sc[g][r] : -FLT_MAX;
            pmax = fmaxf(pmax, v);
        }
    }
    red[g][j] = pmax;
    __syncthreads();
    if (j == 0) {
        float m = red[g][0];
#pragma unroll
        for (int i = 1; i < 16; ++i) m = fmaxf(m, red[g][i]);
        rowmax[g] = m;
    }
    __syncthreads();
    float m = rowmax[g];
    float psum = 0.f;
    for (int t = 0; t < RP / 16; ++t) {
        int r = j + t * 16;
        if (r < RR) {
            float v = mask[b * RR + r] ? sc[g][r] : -FLT_MAX;
            psum += __expf(v - m);
        }
    }
    red[g][j] = psum;
    __syncthreads();
    if (j == 0) {
        float s = 0.f;
#pragma unroll
        for (int i = 0; i < 16; ++i) s += red[g][i];
        rowsum[g] = s;
    }
    __syncthreads();

    // ---- Phase 3: scatter attention weights into [B,L,V] ----
    float inv = 1.0f / rowsum[g];
    size_t outRow = (size_t)(b * LL + l0 + g) * VV;
    for (int t = 0; t < RP / 16; ++t) {
        int r = j + t * 16;
        if (r < RR) {
            float v = mask[b * RR + r] ? sc[g][r] : -FLT_MAX;
            float p = __expf(v - m) * inv;
            out[outRow + ids[b * RR + r]] = p;
        }
    }
}

// ---------------------------------------------------------------------------
extern "C" void kernel_launch(void* const* d_in, const int* in_sizes, int n_in,
                              void* d_out, int out_size, void* d_ws, size_t ws_size,
                              hipStream_t stream) {
    const int*           ids    = (const int*)d_in[0];            // [B,R]
    const float*         embeds = (const float*)d_in[1];          // [B,R,H]
    const unsigned char* mask   = (const unsigned char*)d_in[2];  // [B,R] bool
    const float*         hidden = (const float*)d_in[3];          // [B,L,H]
    const float*         Wq     = (const float*)d_in[5];          // [H,H]
    const float*         bq     = (const float*)d_in[6];          // [H]
    const float*         Wk     = (const float*)d_in[7];          // [H,H]
    const float*         bk     = (const float*)d_in[8];          // [H]
    float*               out    = (float*)d_out;                  // [B,L,V]

    // Workspace: q [B*L, H] then padded k [B*RP, H]
    float* qws = (float*)d_ws;
    float* kws = qws + (size_t)BB * LL * HH;

    // 1) zero-fill output (B*L*V divisible by 4)
    long n4 = (long)BB * LL * VV / 4;
    zero_out_kernel<<<8192, 256, 0, stream>>>((float4*)out, n4);

    // 2) q projection: rows = B*L (no padding): tiles = (2048/16)*(768/16)
    {
        int blocks = (BB * LL / 16) * (HH / 16);
        gemm_proj_kernel<<<blocks, 32, 0, stream>>>(hidden, Wq, bq, qws, LL, LL);
    }
    // 3) k projection into padded rows: rows = B*RP
    {
        int blocks = (BB * RP / 16) * (HH / 16);
        gemm_proj_kernel<<<blocks, 32, 0, stream>>>(embeds, Wk, bk, kws, RP, RR);
    }
    // 4) fused scores + softmax + scatter
    attn_scatter_kernel<<<BB * (LL / 16), 256, 0, stream>>>(qws, kws, ids, mask, out);
}